// Coords2Stress_8229157339458
// MI455X (gfx1250) — compile-verified
//
#include <hip/hip_runtime.h>
#include <hip/hip_bf16.h>

// CDNA5 (gfx1250) wave32 WMMA types
typedef __attribute__((ext_vector_type(2))) float v2f;  // A/B of 16x16x4 f32: 64 vals / 32 lanes
typedef __attribute__((ext_vector_type(8))) float v8f;  // C/D 16x16 f32: 256 vals / 32 lanes

#define LOG2E_F 1.4426950408889634f

__device__ __forceinline__ float fast_sqrtf(float x) {
#if __has_builtin(__builtin_amdgcn_sqrtf)
  return __builtin_amdgcn_sqrtf(x);      // v_sqrt_f32
#else
  return sqrtf(x);
#endif
}

__device__ __forceinline__ float fast_exp_neg(float d) {   // exp(-d)
#if __has_builtin(__builtin_amdgcn_exp2f)
  return __builtin_amdgcn_exp2f(d * (-LOG2E_F));           // v_exp_f32
#else
  return __expf(-d);
#endif
}

// One block = one (batch, 16-row strip). 8 wave32s split the N/16 column tiles.
// Each wave: one V_WMMA_F32_16X16X4_F32 per 16x16 tile produces d^2 directly:
//   A[m]   = (x_m, y_m, z_m, 1)
//   B[:,n] = (-2x_n, -2y_n, -2z_n, |c_n|^2)
//   C[m,n] = |c_m|^2
//   D = A*B + C = |c_m|^2 - 2 c_m.c_n + |c_n|^2 = d2(m,n)
//
// NC: compile-time N (0 = use runtime Nrt). With NC=2048 every row-store uses
// an immediate offset (v*N*4 <= 57KB fits the 24-bit ioffset field).
template <int NC>
__launch_bounds__(256)
__global__ void coords2stress_kernel(const float* __restrict__ coords,
                                     const int* __restrict__ num_atoms,
                                     float* __restrict__ out,
                                     int Nrt) {
  const int N    = NC ? NC : Nrt;
  const int b    = blockIdx.y;
  const int i0   = blockIdx.x << 4;      // row strip start
  const int tid  = threadIdx.x;
  const int lane = tid & 31;
  const int wave = tid >> 5;
  const int lo   = lane & 15;            // column / row-atom index within tile
  const int hi   = lane >> 4;            // half-wave select
  const int nat  = num_atoms[b];
  const int nT   = N >> 4;               // column tiles
  const int NW   = blockDim.x >> 5;      // 8 waves

  __shared__ float s_row[16];
  if (tid < 16) s_row[tid] = 0.0f;
  __syncthreads();

  const float* cb = coords + (size_t)b * 3u * (size_t)N;

  // ---- strip-constant operands (loaded once) ----
  const int   mi  = i0 + lo;                       // row atom handled by this lane
  const float rx  = cb[3 * mi + 0];
  const float ry  = cb[3 * mi + 1];
  const float rz  = cb[3 * mi + 2];
  const float rsq = rx * rx + ry * ry + rz * rz;

  // A-matrix 16x4 f32 layout: VGPR0 = {K0 | K2}, VGPR1 = {K1 | K3} per half-wave
  v2f a;
  a.x = hi ? rz : rx;
  a.y = hi ? 1.0f : ry;

  // C-matrix: lane holds rows M = v + 8*hi, columns = lo. C[m][*] = |c_{i0+m}|^2
  v8f cm;
#pragma unroll
  for (int v = 0; v < 8; ++v)
    cm[v] = __shfl(rsq, v + 8 * hi, 32);

  float acc[8];
#pragma unroll
  for (int v = 0; v < 8; ++v) acc[v] = 0.0f;

  const size_t bOff = (size_t)b * (size_t)N * (size_t)N;

  for (int jt = wave; jt < nT; jt += NW) {
    const int j0 = jt << 4;
    const int nj = j0 + lo;                        // column atom for this lane
    const float bx  = cb[3 * nj + 0];
    const float by  = cb[3 * nj + 1];
    const float bz  = cb[3 * nj + 2];
    const float bsq = bx * bx + by * by + bz * bz;

    // B-matrix 4x16 f32 (symmetric layout to A): VGPR0 = {K0 | K2}, VGPR1 = {K1 | K3}
    v2f bm;
    bm.x = hi ? (-2.0f * bz) : (-2.0f * bx);
    bm.y = hi ? bsq          : (-2.0f * by);

    // d2 tile in one matrix op (8 args: neg_a, A, neg_b, B, c_mod, C, reuse_a, reuse_b)
    v8f d = __builtin_amdgcn_wmma_f32_16x16x4_f32(false, a, false, bm,
                                                  (short)0, cm, false, false);

    float* op = out + bOff + (size_t)(i0 + 8 * hi) * (size_t)N + (size_t)nj;
#pragma unroll
    for (int v = 0; v < 8; ++v) {
      const int gi = i0 + 8 * hi + v;              // global row of this element
      const float d2   = fmaxf(d[v], 0.0f);        // clamp rounding noise
      const float dist = fast_sqrtf(d2);
      float val = -fast_exp_neg(dist);             // -exp(-d / sigma2), sigma2 = 1
      // padded pairs and the (pre-replacement) diagonal are exactly -1
      const bool fixed = (gi >= nat) | (nj >= nat) | (gi == nj);
      val = fixed ? -1.0f : val;
      acc[v] += val;                               // row sum includes diag & pads
      // Unconditional store (diag holds -1 for now; overwritten after the
      // workgroup release fence in __syncthreads below). Immediate offset when NC!=0.
      __builtin_nontemporal_store(val, op + (size_t)v * (size_t)N);
    }
  }

  // reduce row sums across the 16 column-lanes of each half-wave
#pragma unroll
  for (int v = 0; v < 8; ++v) {
#pragma unroll
    for (int m = 1; m < 16; m <<= 1)
      acc[v] += __shfl_xor(acc[v], m, 32);
  }
  if (lo == 0) {
#pragma unroll
    for (int v = 0; v < 8; ++v)
      atomicAdd(&s_row[v + 8 * hi], acc[v]);       // ds_add_f32
  }
  __syncthreads();  // release fence: all loop stores complete before diag write

  // diagonal: -rowsum for valid rows, -1 for padded rows
  if (tid < 16) {
    const int gi = i0 + tid;
    const float dval = (gi < nat) ? -s_row[tid] : -1.0f;
    out[bOff + (size_t)gi * (size_t)N + (size_t)gi] = dval;
  }
}

extern "C" void kernel_launch(void* const* d_in, const int* in_sizes, int n_in,
                              void* d_out, int out_size, void* d_ws, size_t ws_size,
                              hipStream_t stream) {
  const float* coords    = (const float*)d_in[0];
  const int*   num_atoms = (const int*)d_in[1];
  float*       out       = (float*)d_out;

  const int B = in_sizes[1];                 // 16
  const int N = in_sizes[0] / (3 * B);       // 2048

  dim3 grid(N / 16, B);
  if (N == 2048) {
    coords2stress_kernel<2048><<<grid, 256, 0, stream>>>(coords, num_atoms, out, N);
  } else {
    coords2stress_kernel<0><<<grid, 256, 0, stream>>>(coords, num_atoms, out, N);
  }
}